// GVPGraphEmbedding_58256936403257
// MI455X (gfx1250) — compile-verified
//
#include <hip/hip_runtime.h>
#include <hip/hip_bf16.h>
#include <math.h>

// ---------------------------------------------------------------------------
// GVP graph embedding for MI455X (gfx1250, wave32, WMMA).
// All GEMMs use V_WMMA_F32_16X16X4_F32 with guard-free, software-pipelined
// inner loops: M multiples of 16 by construction, K padded to multiples of 4
// (zero-filled), N padded to tile multiples. Fragments loaded as aligned b64.
// Workload is HBM-bound (~200 MB traffic, ~6 GFLOP) so fp32 WMMA is free.
// ---------------------------------------------------------------------------

typedef __attribute__((ext_vector_type(2))) float v2f;
typedef __attribute__((ext_vector_type(8))) float v8f;

constexpr int B_ = 4, L_ = 512, K_ = 30;
constexpr int NBL = B_ * L_;          // 2048 residues
constexpr int NE  = NBL * K_;         // 61440 edges
constexpr int NODE_HS = 1024, NODE_HV = 256;
constexpr int EDGE_HS = 32;

// padded K / N for the WMMA GEMMs
constexpr int KN_VT   = 16;    // node vector GEMM K: 15 -> 16
constexpr int KN_SIN  = 264;   // node scalar GEMM K: 262 -> 264
constexpr int NE_VH   = 64;    // edge vh GEMM N: 56 -> 64 (row stride of vh_e)

// workspace layout (floats)
constexpr size_t OFF_VT_N  = 0;                                  // (NBL*3) x 16
constexpr size_t OFF_VH_N  = OFF_VT_N  + (size_t)NBL*3*KN_VT;    // (NBL*3) x 256
constexpr size_t OFF_SIN_N = OFF_VH_N  + (size_t)NBL*3*256;      // NBL x 264
constexpr size_t OFF_VT_E  = OFF_SIN_N + (size_t)NBL*KN_SIN;     // (NE*3) x 56
constexpr size_t OFF_VH_E  = OFF_VT_E  + (size_t)NE*3*56;        // (NE*3) x 64
constexpr size_t OFF_SIN_E = OFF_VH_E  + (size_t)NE*3*NE_VH;     // NE x 88
constexpr size_t OFF_WHN_P = OFF_SIN_E + (size_t)NE*88;          // 256 x 16
constexpr size_t OFF_WSNW_P= OFF_WHN_P + (size_t)256*KN_VT;      // 1024 x 264
constexpr size_t OFF_WHE_P = OFF_WSNW_P+ (size_t)NODE_HS*KN_SIN; // 64 x 56

// d_out layout (floats)
constexpr size_t OUT_NODE_S = 0;                                 // 2048x1024
constexpr size_t OUT_NODE_V = OUT_NODE_S + (size_t)NBL*NODE_HS;  // 2048x256x3
constexpr size_t OUT_EDGE_S = OUT_NODE_V + (size_t)NBL*NODE_HV*3;// 61440x32
constexpr size_t OUT_EDGE_V = OUT_EDGE_S + (size_t)NE*EDGE_HS;   // 61440x3
constexpr size_t OUT_EI     = OUT_EDGE_V + (size_t)NE*3;         // 2x61440 int

// -------------------------- small vec3 helpers -----------------------------
struct F3 { float x, y, z; };
__device__ __forceinline__ F3 mkf3(float x, float y, float z){ return {x,y,z}; }
__device__ __forceinline__ F3 sub3(F3 a, F3 b){ return {a.x-b.x, a.y-b.y, a.z-b.z}; }
__device__ __forceinline__ F3 add3(F3 a, F3 b){ return {a.x+b.x, a.y+b.y, a.z+b.z}; }
__device__ __forceinline__ float dot3(F3 a, F3 b){ return a.x*b.x + a.y*b.y + a.z*b.z; }
__device__ __forceinline__ F3 cross3(F3 a, F3 b){
  return { a.y*b.z - a.z*b.y, a.z*b.x - a.x*b.z, a.x*b.y - a.y*b.x };
}
__device__ __forceinline__ F3 nrm3(F3 a){  // x / sqrt(clip(|x|^2, 1e-8))
  float inv = 1.f / sqrtf(fmaxf(dot3(a,a), 1e-8f));
  return { a.x*inv, a.y*inv, a.z*inv };
}
__device__ __forceinline__ F3 ldc(const float* c, int b, int res, int atom){
  const float* p = c + ((size_t)((b*L_ + res)*14 + atom))*3;
  return { p[0], p[1], p[2] };
}

// ------------- zero-padded weight copy: dst (Ndst x Kdst) ------------------
__global__ void pad_weight_kernel(const float* __restrict__ src, float* __restrict__ dst,
                                  int Nsrc, int Ksrc, int Ndst, int Kdst)
{
  int t = blockIdx.x*blockDim.x + threadIdx.x;
  if (t >= Ndst*Kdst) return;
  int n = t / Kdst, k = t - n*Kdst;
  dst[t] = (n < Nsrc && k < Ksrc) ? src[(size_t)n*Ksrc + k] : 0.f;
}

// -------------------------- node features ---------------------------------
__global__ void node_feat_kernel(const float* __restrict__ coords,
                                 const unsigned char* __restrict__ pad,
                                 const float* __restrict__ a14,
                                 float* __restrict__ s_in_n,   // NBL x 264, cols 0..5
                                 float* __restrict__ vt_n)     // (NBL*3) x 16
{
  int bl = blockIdx.x*blockDim.x + threadIdx.x;
  if (bl >= NBL) return;
  int b = bl / L_, l = bl - b*L_;

  // dihedrals (3 per residue), backbone chain Xb[m] = coords[b][m/3][m%3]
  float ds[6];
  #pragma unroll
  for (int t=0; t<3; t++){
    int m = 3*l + t;
    float Dv = 0.f;
    if (m >= 1 && m <= 3*L_ - 3){
      F3 p0 = ldc(coords,b,(m-1)/3,(m-1)%3);
      F3 p1 = ldc(coords,b, m   /3, m   %3);
      F3 p2 = ldc(coords,b,(m+1)/3,(m+1)%3);
      F3 p3 = ldc(coords,b,(m+2)/3,(m+2)%3);
      F3 u2 = nrm3(sub3(p1,p0));
      F3 u1 = nrm3(sub3(p2,p1));
      F3 u0 = nrm3(sub3(p3,p2));
      F3 n2 = nrm3(cross3(u2,u1));
      F3 n1 = nrm3(cross3(u1,u0));
      float cd = fminf(fmaxf(dot3(n2,n1), -1.f + 1e-7f), 1.f - 1e-7f);
      float sg = dot3(u2,n1);
      float sgn = (sg > 0.f) ? 1.f : ((sg < 0.f) ? -1.f : 0.f);
      Dv = sgn * acosf(cd);
    }
    ds[t]   = cosf(Dv);
    ds[3+t] = sinf(Dv);
  }
  #pragma unroll
  for (int c=0; c<6; c++) s_in_n[(size_t)bl*KN_SIN + c] = ds[c];
  s_in_n[(size_t)bl*KN_SIN + 262] = 0.f;      // K padding
  s_in_n[(size_t)bl*KN_SIN + 263] = 0.f;

  // 15 vector channels: fwd, bwd, 13 frame atoms
  F3 nv[15];
  F3 ca = ldc(coords,b,l,1);
  nv[0] = (l < L_-1) ? nrm3(sub3(ldc(coords,b,l+1,1), ca)) : mkf3(0,0,0);
  nv[1] = (l > 0)    ? nrm3(sub3(ldc(coords,b,l-1,1), ca)) : mkf3(0,0,0);
  const int AP[13] = {0,2,3,4,5,6,7,8,9,10,11,12,13};
  #pragma unroll
  for (int i=0; i<13; i++) nv[2+i] = nrm3(sub3(ca, ldc(coords,b,l,AP[i])));

  bool no_cb = (a14[(size_t)bl*14 + 4] == 0.f) && (pad[bl] == 0);
  if (no_cb){
    F3 n  = nrm3(sub3(ldc(coords,b,l,0), ca));
    F3 c  = nrm3(sub3(ldc(coords,b,l,2), ca));
    F3 bi = nrm3(add3(c,n));
    F3 pp = nrm3(cross3(c,n));
    float c1 = sqrtf(1.f/3.f), c2 = sqrtf(2.f/3.f);
    nv[5] = mkf3(-bi.x*c1 - pp.x*c2, -bi.y*c1 - pp.y*c2, -bi.z*c1 - pp.z*c2);
  }
  #pragma unroll
  for (int j=0; j<15; j++){
    vt_n[((size_t)bl*3 + 0)*KN_VT + j] = nv[j].x;
    vt_n[((size_t)bl*3 + 1)*KN_VT + j] = nv[j].y;
    vt_n[((size_t)bl*3 + 2)*KN_VT + j] = nv[j].z;
  }
  vt_n[((size_t)bl*3 + 0)*KN_VT + 15] = 0.f;  // K padding
  vt_n[((size_t)bl*3 + 1)*KN_VT + 15] = 0.f;
  vt_n[((size_t)bl*3 + 2)*KN_VT + 15] = 0.f;
}

// -------------------------- edge features ---------------------------------
__global__ void edge_feat_kernel(const float* __restrict__ coords,
                                 const int* __restrict__ Eidx,
                                 float* __restrict__ s_in_e,   // NE x 88, cols 0..31
                                 float* __restrict__ vt_e)     // (NE*3) x 56
{
  int e = blockIdx.x*blockDim.x + threadIdx.x;
  if (e >= NE) return;
  int b = e / (L_*K_);
  int rem = e - b*(L_*K_);
  int l = rem / K_;
  int idx = Eidx[e];

  F3 xs = ldc(coords,b,l,1), xd = ldc(coords,b,idx,1);
  F3 dv = sub3(xs,xd);
  float dist = sqrtf(dot3(dv,dv) + 1e-8f);

  float* srow = s_in_e + (size_t)e*88;
  #pragma unroll
  for (int i=0; i<16; i++){
    float c = 20.f * (float)i / 15.f;
    float z = (dist - c) * 0.8f;        // / (20/16)
    srow[i] = expf(-z*z);
  }
  float d = (float)(l - idx);
  #pragma unroll
  for (int i=0; i<8; i++){
    float fr = expf(-(float)(2*i) * (9.210340371976184f/16.f));  // ln(1e4)/NPE
    float a = d * fr;
    srow[16+i] = cosf(a);
    srow[24+i] = sinf(a);
  }

  const int BBA[4] = {1,0,2,3};
  F3 bb[4];
  #pragma unroll
  for (int i=0;i<4;i++) bb[i] = ldc(coords,b,l,BBA[i]);
  for (int a4=0; a4<4; a4++){
    for (int a14=0; a14<14; a14++){
      F3 g = ldc(coords,b,idx,a14);
      F3 v = nrm3(sub3(bb[a4], g));     // double-normalize is idempotent
      int vi = a4*14 + a14;
      vt_e[((size_t)e*3 + 0)*56 + vi] = v.x;
      vt_e[((size_t)e*3 + 1)*56 + vi] = v.y;
      vt_e[((size_t)e*3 + 2)*56 + vi] = v.z;
    }
  }
}

// ------------- vn = ||vh||_xyz into concat GEMM input rows -----------------
__global__ void vn_concat_kernel(const float* __restrict__ vh, float* __restrict__ s_in,
                                 int nrows, int nvch, int vstride, int stride, int coloff)
{
  int t = blockIdx.x*blockDim.x + threadIdx.x;
  if (t >= nrows*nvch) return;
  int row = t / nvch, c = t - row*nvch;
  const float* p = vh + (size_t)row*3*vstride;
  float x = p[c], y = p[vstride + c], z = p[2*vstride + c];
  s_in[(size_t)row*stride + coloff + c] = sqrtf(fmaxf(x*x + y*y + z*z, 1e-8f));
}

// -------------------- WMMA fp32 GEMM: D = A @ W^T (+bias / +=) -------------
// Preconditions: M % 16 == 0, N % (16*NTILE) == 0, K % 4 == 0, K >= 4
// (zero-padded). A: M x K row-major.  W: N x K row-major (B^T).
// Software-pipelined: iteration k+1 fragments are loaded before iteration k's
// WMMAs issue, so loads hide under matrix-pipe work.
// vec3_store: D[(m/3)*N*3 + n*3 + m%3]  (for (rows, ch, xyz) vector outputs)
template<int NTILE>
__global__ void gemm_wmma_f32(const float* __restrict__ A,
                              const float* __restrict__ W,
                              const float* __restrict__ bias,
                              float* __restrict__ D,
                              int M, int N, int K,
                              int accumulate, int vec3_store)
{
  const int lane = threadIdx.x & 31;
  const int wave = threadIdx.x >> 5;
  const int wpb  = blockDim.x >> 5;
  const int tilesN = N / (16*NTILE);
  const int tileId = blockIdx.x * wpb + wave;
  if (tileId >= (M >> 4) * tilesN) return;
  const int tm = tileId / tilesN;
  const int tn = tileId - tm*tilesN;
  const int m0  = tm << 4;
  const int n0b = tn * (16*NTILE);

  const int half = lane >> 4;     // 0/1
  const int r16  = lane & 15;
  const int kb   = half << 1;     // A/B: K pair base (0 or 2)
  const int mb   = half << 3;     // C/D: M half base (0 or 8)

  v8f acc[NTILE];
  #pragma unroll
  for (int t=0; t<NTILE; t++){
    if (accumulate){
      #pragma unroll
      for (int i=0; i<8; i++){
        int m = m0 + mb + i, n = n0b + t*16 + r16;
        acc[t][i] = vec3_store ? D[(size_t)(m/3)*N*3 + (size_t)n*3 + (m%3)]
                               : D[(size_t)m*N + n];
      }
    } else {
      #pragma unroll
      for (int i=0; i<8; i++) acc[t][i] = 0.f;
    }
  }

  const float* Ap = A + (size_t)(m0 + r16)*K + kb;
  const float* Wp = W + (size_t)(n0b + r16)*K + kb;

  // prologue: fragments for k0 = 0
  v2f afr = *(const v2f*)(Ap);
  v2f bfr[NTILE];
  #pragma unroll
  for (int t=0; t<NTILE; t++) bfr[t] = *(const v2f*)(Wp + (size_t)t*16*K);

  for (int k0 = 4; k0 < K; k0 += 4){
    // prefetch next iteration's fragments
    v2f afr_n = *(const v2f*)(Ap + k0);
    v2f bfr_n[NTILE];
    #pragma unroll
    for (int t=0; t<NTILE; t++) bfr_n[t] = *(const v2f*)(Wp + (size_t)t*16*K + k0);
    // consume current fragments on the matrix pipe
    #pragma unroll
    for (int t=0; t<NTILE; t++)
      acc[t] = __builtin_amdgcn_wmma_f32_16x16x4_f32(
                 false, afr, false, bfr[t], (short)0, acc[t], false, false);
    afr = afr_n;
    #pragma unroll
    for (int t=0; t<NTILE; t++) bfr[t] = bfr_n[t];
  }
  // epilogue: last k-step
  #pragma unroll
  for (int t=0; t<NTILE; t++)
    acc[t] = __builtin_amdgcn_wmma_f32_16x16x4_f32(
               false, afr, false, bfr[t], (short)0, acc[t], false, false);

  #pragma unroll
  for (int t=0; t<NTILE; t++){
    int n = n0b + t*16 + r16;
    float bv = (bias != nullptr) ? bias[n] : 0.f;
    #pragma unroll
    for (int i=0; i<8; i++){
      int m = m0 + mb + i;
      float val = acc[t][i] + bv;
      if (vec3_store) D[(size_t)(m/3)*N*3 + (size_t)n*3 + (m%3)] = val;
      else            D[(size_t)m*N + n] = val;
    }
  }
}

// ---------------- edge v_out: (NE*3, 64-strided) . wv_e(56) -> edge_v ------
__global__ void edge_vout_kernel(const float* __restrict__ vh_e,
                                 const float* __restrict__ wv_e,
                                 float* __restrict__ edge_v)
{
  int t = blockIdx.x*blockDim.x + threadIdx.x;
  if (t >= NE*3) return;
  const float* src = vh_e + (size_t)t*NE_VH;
  float acc = 0.f;
  #pragma unroll
  for (int j=0; j<56; j++) acc += src[j]*wv_e[j];
  edge_v[t] = acc;
}

// ---------------- tuple_ln for nodes (1 block / residue) -------------------
__global__ void tuple_ln_node_kernel(float* __restrict__ s, float* __restrict__ v,
                                     const float* __restrict__ g,
                                     const float* __restrict__ beta)
{
  const int row = blockIdx.x, tid = threadIdx.x;
  float* srow = s + (size_t)row*NODE_HS;
  float* vrow = v + (size_t)row*NODE_HV*3;
  __shared__ float sd[256];

  float sv[4]; float ls = 0.f, lq = 0.f;
  #pragma unroll
  for (int j=0; j<4; j++){ float x = srow[tid + 256*j]; sv[j]=x; ls+=x; lq+=x*x; }
  sd[tid] = ls; __syncthreads();
  for (int st=128; st>0; st>>=1){ if (tid<st) sd[tid]+=sd[tid+st]; __syncthreads(); }
  float mu = sd[0] * (1.f/NODE_HS); __syncthreads();
  sd[tid] = lq; __syncthreads();
  for (int st=128; st>0; st>>=1){ if (tid<st) sd[tid]+=sd[tid+st]; __syncthreads(); }
  float var = sd[0]*(1.f/NODE_HS) - mu*mu; __syncthreads();
  float inv = 1.f/sqrtf(var + 1e-5f);
  #pragma unroll
  for (int j=0; j<4; j++){
    int c = tid + 256*j;
    srow[c] = (sv[j]-mu)*inv*g[c] + beta[c];
  }
  // vector half: one vector channel per thread (NODE_HV == blockDim == 256)
  float vx = vrow[tid*3+0], vy = vrow[tid*3+1], vz = vrow[tid*3+2];
  float vn = fmaxf(vx*vx + vy*vy + vz*vz, 1e-4f);
  float m  = (vn > 2e-4f) ? 1.f : 0.f;
  sd[tid] = vn*m; __syncthreads();
  for (int st=128; st>0; st>>=1){ if (tid<st) sd[tid]+=sd[tid+st]; __syncthreads(); }
  float sum_vnm = sd[0]; __syncthreads();
  sd[tid] = m; __syncthreads();
  for (int st=128; st>0; st>>=1){ if (tid<st) sd[tid]+=sd[tid+st]; __syncthreads(); }
  float sum_m = sd[0];
  float vnorm = sqrtf(sum_vnm/(1e-4f + sum_m) + 1e-4f);
  float sc = m / vnorm;
  vrow[tid*3+0] = vx*sc; vrow[tid*3+1] = vy*sc; vrow[tid*3+2] = vz*sc;
}

// ---------------- tuple_ln for edges (1 wave / edge) -----------------------
__global__ void tuple_ln_edge_kernel(float* __restrict__ es, float* __restrict__ ev,
                                     const float* __restrict__ g,
                                     const float* __restrict__ beta)
{
  int gid = blockIdx.x*blockDim.x + threadIdx.x;
  int row = gid >> 5, lane = gid & 31;
  if (row >= NE) return;
  float* srow = es + (size_t)row*EDGE_HS;
  float x = srow[lane];
  float s = x, q = x*x;
  #pragma unroll
  for (int off=16; off>0; off>>=1){
    s += __shfl_xor(s, off, 32);
    q += __shfl_xor(q, off, 32);
  }
  float mu  = s * (1.f/32.f);
  float var = q * (1.f/32.f) - mu*mu;
  float inv = 1.f/sqrtf(var + 1e-5f);
  srow[lane] = (x-mu)*inv*g[lane] + beta[lane];
  if (lane == 0){
    float* vr = ev + (size_t)row*3;
    float vx=vr[0], vy=vr[1], vz=vr[2];
    float vn = fmaxf(vx*vx + vy*vy + vz*vz, 1e-4f);
    float m  = (vn > 2e-4f) ? 1.f : 0.f;
    float vnorm = sqrtf(vn*m/(1e-4f + m) + 1e-4f);
    float sc = m / vnorm;
    vr[0]=vx*sc; vr[1]=vy*sc; vr[2]=vz*sc;
  }
}

// -------- final: edge mask multiply + edge index construction --------------
__global__ void mask_ei_kernel(const int* __restrict__ Eidx,
                               const unsigned char* __restrict__ pad,
                               float* __restrict__ edge_s, float* __restrict__ edge_v,
                               int* __restrict__ ei)
{
  int e = blockIdx.x*blockDim.x + threadIdx.x;
  if (e >= NE) return;
  int b = e / (L_*K_);
  int rem = e - b*(L_*K_);
  int l = rem / K_;
  int idx = Eidx[e];
  bool valid = (pad[b*L_ + l] == 0) && (pad[b*L_ + idx] == 0);
  float mf = valid ? 1.f : 0.f;
  ei[e]      = valid ? (l   + b*L_) : -1;
  ei[NE + e] = valid ? (idx + b*L_) : -1;
  #pragma unroll
  for (int c=0; c<EDGE_HS; c++) edge_s[(size_t)e*EDGE_HS + c] *= mf;
  edge_v[(size_t)e*3+0] *= mf; edge_v[(size_t)e*3+1] *= mf; edge_v[(size_t)e*3+2] *= mf;
}

// ---------------------------------------------------------------------------
extern "C" void kernel_launch(void* const* d_in, const int* in_sizes, int n_in,
                              void* d_out, int out_size, void* d_ws, size_t ws_size,
                              hipStream_t stream)
{
  const float* coords  = (const float*)d_in[0];
  const int*   Eidx    = (const int*)  d_in[2];
  const float* mp_node = (const float*)d_in[3];
  const float* mp_edge = (const float*)d_in[4];
  const unsigned char* pad = (const unsigned char*)d_in[5];
  const float* a14     = (const float*)d_in[6];
  const float* wh_n    = (const float*)d_in[7];
  const float* ws_n_w  = (const float*)d_in[8];
  const float* ws_n_b  = (const float*)d_in[9];
  const float* wv_n    = (const float*)d_in[10];
  const float* ln_n_g  = (const float*)d_in[11];
  const float* ln_n_b  = (const float*)d_in[12];
  const float* wh_e    = (const float*)d_in[13];
  const float* ws_e_w  = (const float*)d_in[14];
  const float* ws_e_b  = (const float*)d_in[15];
  const float* wv_e    = (const float*)d_in[16];
  const float* ln_e_g  = (const float*)d_in[17];
  const float* ln_e_b  = (const float*)d_in[18];
  const float* w_mn    = (const float*)d_in[19];
  const float* b_mn    = (const float*)d_in[20];
  const float* w_me    = (const float*)d_in[21];
  const float* b_me    = (const float*)d_in[22];

  float* ws = (float*)d_ws;
  float* vt_n    = ws + OFF_VT_N;
  float* vh_n    = ws + OFF_VH_N;
  float* s_in_n  = ws + OFF_SIN_N;
  float* vt_e    = ws + OFF_VT_E;
  float* vh_e    = ws + OFF_VH_E;
  float* s_in_e  = ws + OFF_SIN_E;
  float* whn_p   = ws + OFF_WHN_P;
  float* wsnw_p  = ws + OFF_WSNW_P;
  float* whe_p   = ws + OFF_WHE_P;

  float* out    = (float*)d_out;
  float* node_s = out + OUT_NODE_S;
  float* node_v = out + OUT_NODE_V;
  float* edge_s = out + OUT_EDGE_S;
  float* edge_v = out + OUT_EDGE_V;
  int*   ei     = (int*)(out + OUT_EI);

  auto gemm = [&](int ntile, const float* A, const float* W, const float* bias,
                  float* D, int M, int N, int K, int accum, int v3){
    int tiles  = (M/16) * (N/(16*ntile));
    int blocks = (tiles + 7) / 8;               // 8 waves / 256-thread block
    if (ntile == 4)
      gemm_wmma_f32<4><<<blocks, 256, 0, stream>>>(A,W,bias,D,M,N,K,accum,v3);
    else
      gemm_wmma_f32<2><<<blocks, 256, 0, stream>>>(A,W,bias,D,M,N,K,accum,v3);
  };

  // ---- pad weights that need K/N rounding (tiny, one-time per launch) ----
  pad_weight_kernel<<<(256*KN_VT+255)/256, 256, 0, stream>>>(wh_n, whn_p, 256, 15, 256, KN_VT);
  pad_weight_kernel<<<(NODE_HS*KN_SIN+255)/256, 256, 0, stream>>>(ws_n_w, wsnw_p, NODE_HS, 262, NODE_HS, KN_SIN);
  pad_weight_kernel<<<(NE_VH*56+255)/256, 256, 0, stream>>>(wh_e, whe_p, 56, 56, NE_VH, 56);

  // ---- features ----
  node_feat_kernel<<<(NBL+255)/256, 256, 0, stream>>>(coords, pad, a14, s_in_n, vt_n);
  edge_feat_kernel<<<(NE+255)/256, 256, 0, stream>>>(coords, Eidx, s_in_e, vt_e);

  // ---- node path ----
  gemm(4, vt_n, whn_p, nullptr, vh_n, NBL*3, NODE_HV, KN_VT, 0, 0);       // vh
  {
    int n = NBL*NODE_HV;
    vn_concat_kernel<<<(n+255)/256, 256, 0, stream>>>(vh_n, s_in_n, NBL, NODE_HV, NODE_HV, KN_SIN, 6);
  }
  gemm(4, s_in_n, wsnw_p, ws_n_b, node_s, NBL, NODE_HS, KN_SIN, 0, 0);    // s
  gemm(4, vh_n, wv_n, nullptr, node_v, NBL*3, NODE_HV, NODE_HV, 0, 1);    // v_out
  tuple_ln_node_kernel<<<NBL, 256, 0, stream>>>(node_s, node_v, ln_n_g, ln_n_b);
  gemm(4, mp_node, w_mn, b_mn, node_s, NBL, NODE_HS, 128, 1, 0);          // += mpnn

  // ---- edge path ----
  gemm(4, vt_e, whe_p, nullptr, vh_e, NE*3, NE_VH, 56, 0, 0);             // vh_e (N padded 64)
  {
    int n = NE*56;
    vn_concat_kernel<<<(n+255)/256, 256, 0, stream>>>(vh_e, s_in_e, NE, 56, NE_VH, 88, 32);
  }
  gemm(2, s_in_e, ws_e_w, ws_e_b, edge_s, NE, EDGE_HS, 88, 0, 0);         // e_s
  edge_vout_kernel<<<(NE*3+255)/256, 256, 0, stream>>>(vh_e, wv_e, edge_v);
  tuple_ln_edge_kernel<<<(NE*32+255)/256, 256, 0, stream>>>(edge_s, edge_v, ln_e_g, ln_e_b);
  gemm(2, mp_edge, w_me, b_me, edge_s, NE, EDGE_HS, 512, 1, 0);           // += mpnn_edge
  mask_ei_kernel<<<(NE+255)/256, 256, 0, stream>>>(Eidx, pad, edge_s, edge_v, ei);
}